// RENBase_44229573214832
// MI455X (gfx1250) — compile-verified
//
#include <hip/hip_runtime.h>

// ---------------------------------------------------------------------------
// Fused REN forward for gfx1250 (MI455X), fp32 via V_WMMA_F32_16X16X4_F32.
//   b  = x@C1^T + u@D12^T + bv          [B,128]
//   w  = forward-substitution relu solve (blocked 16, cross-block via WMMA)
//   x1 = x@A^T + w@B1^T + u@B2^T + bx   [B,256]
//   y  = x@C2^T + w@D21^T + u@D22^T+ by [B,128]
// One block = 128 batch rows; Z=[x|u|w] tile lives in 258KB of LDS.
// 2M x 2N register blocking: each wave owns a 32-row M-pair and an N-half,
// so every (A,B) fragment pair feeds 4 WMMAs (halves LDS+L2 bytes per WMMA).
// ---------------------------------------------------------------------------

typedef float v2f __attribute__((ext_vector_type(2)));
typedef float v8f __attribute__((ext_vector_type(8)));

#define BATCH  32768
#define NU_    128
#define NX_    256
#define NV_    128
#define NY_    128
#define TB     128                // batch rows per block
#define ZSTR   516                // padded LDS row stride (516 % 64 == 4 -> conflict-free)
#define ZCOL_U NX_                // 256
#define ZCOL_W (NX_ + NU_)        // 384
#define NTHREADS 256

#define WMMA_F32(a, b, c) \
  __builtin_amdgcn_wmma_f32_16x16x4_f32(false, (a), false, (b), (short)0, (c), false, false)

__device__ __forceinline__ v8f zero8() {
  v8f z = {0.f, 0.f, 0.f, 0.f, 0.f, 0.f, 0.f, 0.f};
  return z;
}

struct Acc22 { v8f c00, c01, c10, c11; };

// 2M x 2N K-segment: c(mi,nj) += Z[mtile_i, k0:k0+kc] * Wseg[ntile_j]^T
// z0/z1: per-lane A-fragment base for M-tiles m0, m0+16 (LDS)
// w0/w1: per-lane B-fragment base for N-tiles n0, n0+16 (global, L2-resident)
__device__ __forceinline__ void seg_mm22(Acc22& ac,
                                         const float* __restrict__ z0,
                                         const float* __restrict__ z1,
                                         const float* __restrict__ w0,
                                         const float* __restrict__ w1,
                                         int kcount) {
#pragma unroll 4
  for (int k = 0; k < kcount; k += 4) {
    v2f a0 = *(const v2f*)(z0 + k);
    v2f a1 = *(const v2f*)(z1 + k);
    v2f b0 = *(const v2f*)(w0 + k);
    v2f b1 = *(const v2f*)(w1 + k);
    ac.c00 = WMMA_F32(a0, b0, ac.c00);
    ac.c01 = WMMA_F32(a0, b1, ac.c01);
    ac.c10 = WMMA_F32(a1, b0, ac.c10);
    ac.c11 = WMMA_F32(a1, b1, ac.c11);
  }
}

// 2M x 1N K-segment (used by the triangular-solve cross-block GEMM).
__device__ __forceinline__ void seg_mm21(v8f& c0, v8f& c1,
                                         const float* __restrict__ z0,
                                         const float* __restrict__ z1,
                                         const float* __restrict__ w0,
                                         int kcount) {
#pragma unroll 4
  for (int k = 0; k < kcount; k += 4) {
    v2f a0 = *(const v2f*)(z0 + k);
    v2f a1 = *(const v2f*)(z1 + k);
    v2f b0 = *(const v2f*)(w0 + k);
    c0 = WMMA_F32(a0, b0, c0);
    c1 = WMMA_F32(a1, b0, c1);
  }
}

__global__ __launch_bounds__(NTHREADS, 1)
void ren_fused_wmma(const float* __restrict__ x,  const float* __restrict__ u,
                    const float* __restrict__ A,  const float* __restrict__ B1,
                    const float* __restrict__ B2, const float* __restrict__ C1,
                    const float* __restrict__ C2, const float* __restrict__ D11,
                    const float* __restrict__ D12,const float* __restrict__ D21,
                    const float* __restrict__ D22,const float* __restrict__ bx,
                    const float* __restrict__ bv, const float* __restrict__ by,
                    float* __restrict__ outX1,    float* __restrict__ outY) {
  extern __shared__ float lds[];  // [TB][ZSTR] = 258 KB

  const int tid  = threadIdx.x;
  const int lane = tid & 31;          // wave32
  const int wave = tid >> 5;          // 0..7
  const int kh   = lane >> 4;         // K-half selector of the fragment
  const int l16  = lane & 15;
  const int m0   = (wave & 3) * 32;   // M-pair: rows m0..m0+31 (two 16-row tiles)
  const int nh   = wave >> 2;         // N-half selector
  const size_t rowbase = (size_t)blockIdx.x * TB;

  // ---- Phase A: stage x,u tiles into LDS (coalesced b128) ------------------
  {
    const float4* xg = (const float4*)(x + rowbase * NX_);
    for (int i = tid; i < TB * NX_ / 4; i += NTHREADS) {
      int r = i >> 6, c4 = i & 63;                     // 64 float4 per row
      *(float4*)&lds[r * ZSTR + c4 * 4] = xg[r * (NX_ / 4) + c4];
    }
    const float4* ug = (const float4*)(u + rowbase * NU_);
    for (int i = tid; i < TB * NU_ / 4; i += NTHREADS) {
      int r = i >> 5, c4 = i & 31;                     // 32 float4 per row
      *(float4*)&lds[r * ZSTR + ZCOL_U + c4 * 4] = ug[r * (NU_ / 4) + c4];
    }
  }
  __syncthreads();

  // Per-lane A-fragment base pointers for the two M-tiles.
  const float* zA0 = &lds[(m0 + l16) * ZSTR + 2 * kh];
  const float* zA1 = zA0 + 16 * ZSTR;

  // ---- Phase B: b = x@C1^T + u@D12^T + bv  -> LDS w-region -----------------
  for (int j = 0; j < 2; ++j) {                 // 2 dual-N iterations: 64 cols
    const int n0 = nh * 64 + j * 32;
    Acc22 ac; ac.c00 = zero8(); ac.c01 = zero8(); ac.c10 = zero8(); ac.c11 = zero8();
    seg_mm22(ac, zA0, zA1,
             C1 + (n0 + l16) * NX_ + 2 * kh,  C1 + (n0 + 16 + l16) * NX_ + 2 * kh, NX_);
    seg_mm22(ac, zA0 + ZCOL_U, zA1 + ZCOL_U,
             D12 + (n0 + l16) * NU_ + 2 * kh, D12 + (n0 + 16 + l16) * NU_ + 2 * kh, NU_);
    const float bv0 = bv[n0 + l16], bv1 = bv[n0 + 16 + l16];
#pragma unroll
    for (int i = 0; i < 8; ++i) {
      const int r0 = (m0 + i + 8 * kh) * ZSTR + ZCOL_W;
      const int r1 = (m0 + 16 + i + 8 * kh) * ZSTR + ZCOL_W;
      lds[r0 + n0 + l16]      = ac.c00[i] + bv0;
      lds[r0 + n0 + 16 + l16] = ac.c01[i] + bv1;
      lds[r1 + n0 + l16]      = ac.c10[i] + bv0;
      lds[r1 + n0 + 16 + l16] = ac.c11[i] + bv1;
    }
  }
  __syncthreads();

  // ---- Phase C: blocked forward-substitution  w = relu(D11_low @ w + b) ----
  for (int kb = 0; kb < NV_ / 16; ++kb) {
    const int i0 = kb * 16;
    if (kb > 0 && nh == 0) {   // wave-uniform: EXEC all-ones inside for WMMA
      // Cross-block: p = b' + w[:, :i0] @ D11[i0:i0+16, :i0]^T   (WMMA, dual-M)
      v8f c0, c1;
#pragma unroll
      for (int i = 0; i < 8; ++i) {
        c0[i] = lds[(m0 + i + 8 * kh) * ZSTR + ZCOL_W + i0 + l16];
        c1[i] = lds[(m0 + 16 + i + 8 * kh) * ZSTR + ZCOL_W + i0 + l16];
      }
      seg_mm21(c0, c1, zA0 + ZCOL_W, zA1 + ZCOL_W,
               D11 + (i0 + l16) * NV_ + 2 * kh, i0);
#pragma unroll
      for (int i = 0; i < 8; ++i) {
        lds[(m0 + i + 8 * kh) * ZSTR + ZCOL_W + i0 + l16]      = c0[i];
        lds[(m0 + 16 + i + 8 * kh) * ZSTR + ZCOL_W + i0 + l16] = c1[i];
      }
    }
    __syncthreads();
    // Intra-block: 16 sequential strictly-lower steps, one thread per row.
    if (tid < TB) {
      float wv[16];
      float* zr = &lds[tid * ZSTR + ZCOL_W + i0];
#pragma unroll
      for (int n = 0; n < 16; ++n) {
        float s = zr[n];
#pragma unroll
        for (int m = 0; m < 16; ++m)
          if (m < n) s += D11[(i0 + n) * NV_ + i0 + m] * wv[m];
        wv[n] = fmaxf(s, 0.f);
        zr[n] = wv[n];
      }
    }
    __syncthreads();
  }

  // ---- Phase D1: x1 = Z @ [A|B2|B1]^T + bx ---------------------------------
  for (int j = 0; j < 4; ++j) {                 // 4 dual-N iterations: 128 cols
    const int n0 = nh * 128 + j * 32;
    Acc22 ac; ac.c00 = zero8(); ac.c01 = zero8(); ac.c10 = zero8(); ac.c11 = zero8();
    seg_mm22(ac, zA0, zA1,
             A + (n0 + l16) * NX_ + 2 * kh,   A + (n0 + 16 + l16) * NX_ + 2 * kh, NX_);
    seg_mm22(ac, zA0 + ZCOL_U, zA1 + ZCOL_U,
             B2 + (n0 + l16) * NU_ + 2 * kh,  B2 + (n0 + 16 + l16) * NU_ + 2 * kh, NU_);
    seg_mm22(ac, zA0 + ZCOL_W, zA1 + ZCOL_W,
             B1 + (n0 + l16) * NV_ + 2 * kh,  B1 + (n0 + 16 + l16) * NV_ + 2 * kh, NV_);
    const float bx0 = bx[n0 + l16], bx1 = bx[n0 + 16 + l16];
#pragma unroll
    for (int i = 0; i < 8; ++i) {
      float* o0 = outX1 + (rowbase + m0 + i + 8 * kh) * NX_;
      float* o1 = outX1 + (rowbase + m0 + 16 + i + 8 * kh) * NX_;
      o0[n0 + l16]      = ac.c00[i] + bx0;
      o0[n0 + 16 + l16] = ac.c01[i] + bx1;
      o1[n0 + l16]      = ac.c10[i] + bx0;
      o1[n0 + 16 + l16] = ac.c11[i] + bx1;
    }
  }

  // ---- Phase D2: y = Z @ [C2|D22|D21]^T + by -------------------------------
  for (int j = 0; j < 2; ++j) {                 // 2 dual-N iterations: 64 cols
    const int n0 = nh * 64 + j * 32;
    Acc22 ac; ac.c00 = zero8(); ac.c01 = zero8(); ac.c10 = zero8(); ac.c11 = zero8();
    seg_mm22(ac, zA0, zA1,
             C2 + (n0 + l16) * NX_ + 2 * kh,   C2 + (n0 + 16 + l16) * NX_ + 2 * kh, NX_);
    seg_mm22(ac, zA0 + ZCOL_U, zA1 + ZCOL_U,
             D22 + (n0 + l16) * NU_ + 2 * kh,  D22 + (n0 + 16 + l16) * NU_ + 2 * kh, NU_);
    seg_mm22(ac, zA0 + ZCOL_W, zA1 + ZCOL_W,
             D21 + (n0 + l16) * NV_ + 2 * kh,  D21 + (n0 + 16 + l16) * NV_ + 2 * kh, NV_);
    const float by0 = by[n0 + l16], by1 = by[n0 + 16 + l16];
#pragma unroll
    for (int i = 0; i < 8; ++i) {
      float* o0 = outY + (rowbase + m0 + i + 8 * kh) * NY_;
      float* o1 = outY + (rowbase + m0 + 16 + i + 8 * kh) * NY_;
      o0[n0 + l16]      = ac.c00[i] + by0;
      o0[n0 + 16 + l16] = ac.c01[i] + by1;
      o1[n0 + l16]      = ac.c10[i] + by0;
      o1[n0 + 16 + l16] = ac.c11[i] + by1;
    }
  }
}

extern "C" void kernel_launch(void* const* d_in, const int* in_sizes, int n_in,
                              void* d_out, int out_size, void* d_ws, size_t ws_size,
                              hipStream_t stream) {
  (void)in_sizes; (void)n_in; (void)out_size; (void)d_ws; (void)ws_size;
  const float* x   = (const float*)d_in[0];
  const float* u   = (const float*)d_in[1];
  const float* A   = (const float*)d_in[2];
  const float* B1  = (const float*)d_in[3];
  const float* B2  = (const float*)d_in[4];
  const float* C1  = (const float*)d_in[5];
  const float* C2  = (const float*)d_in[6];
  const float* D11 = (const float*)d_in[7];
  const float* D12 = (const float*)d_in[8];
  const float* D21 = (const float*)d_in[9];
  const float* D22 = (const float*)d_in[10];
  const float* bx  = (const float*)d_in[11];
  const float* bv  = (const float*)d_in[12];
  const float* by  = (const float*)d_in[13];

  float* outX1 = (float*)d_out;
  float* outY  = outX1 + (size_t)BATCH * NX_;

  const size_t ldsBytes = (size_t)TB * ZSTR * sizeof(float);  // 264192 B
  (void)hipFuncSetAttribute((const void*)ren_fused_wmma,
                            hipFuncAttributeMaxDynamicSharedMemorySize,
                            (int)ldsBytes);

  dim3 grid(BATCH / TB);   // 256 blocks
  dim3 block(NTHREADS);    // 8 wave32 waves
  ren_fused_wmma<<<grid, block, ldsBytes, stream>>>(
      x, u, A, B1, B2, C1, C2, D11, D12, D21, D22, bx, bv, by, outX1, outY);
}